// RoITarget_68796786147718
// MI455X (gfx1250) — compile-verified
//
#include <hip/hip_runtime.h>
#include <hip/hip_bf16.h>
#include <stdint.h>

#define N_GT   1024
#define BLOCK  128

#define AS1 __attribute__((address_space(1)))
#define AS3 __attribute__((address_space(3)))

// ---------------------------------------------------------------------------
// gfx1250 async global->LDS copy (tracked by ASYNCcnt). Builtin if present,
// otherwise raw CDNA5 mnemonic via inline asm.
// Builtin signature (from clang diagnostic): first param is AS1 int*,
// so pass typed addrspace pointers.
// ---------------------------------------------------------------------------
__device__ __forceinline__ void async_g2l_b32(void* lds_dst, const void* gsrc) {
#if __has_builtin(__builtin_amdgcn_global_load_async_to_lds_b32)
    __builtin_amdgcn_global_load_async_to_lds_b32(
        (AS1 int*)(uintptr_t)gsrc,    // global src (global addr bits == flat bits)
        (AS3 int*)lds_dst,            // LDS dst (addrspacecast strips aperture)
        0, 0);
#else
    unsigned lds_off = (unsigned)(uintptr_t)(AS3 void*)lds_dst;
    asm volatile("global_load_async_to_lds_b32 %0, %1, off"
                 :: "v"(lds_off), "v"(gsrc) : "memory");
#endif
}

__device__ __forceinline__ void wait_async0() {
#if __has_builtin(__builtin_amdgcn_s_wait_asynccnt)
    __builtin_amdgcn_s_wait_asynccnt(0);
#else
    asm volatile("s_wait_asynccnt 0x0" ::: "memory");
#endif
}

// ---------------------------------------------------------------------------
// One thread per ROI. GT table staged to LDS via async-to-LDS DMA path.
//   s_gt4[g]  = {x1, y1, x2, y2}
//   s_meta[g] = {area_g, label, batch_id_bits, unused}
// ---------------------------------------------------------------------------
__global__ __launch_bounds__(BLOCK) void RoITarget_kernel(
    const float* __restrict__ rois,        // (N_ROIS, 5)
    const int*   __restrict__ roi_binds,   // (N_ROIS)
    const float* __restrict__ gt_boxes,    // (N_GT, 5)
    const int*   __restrict__ gt_binds,    // (N_GT)
    float* __restrict__ out_labels,        // (N_ROIS)
    float* __restrict__ out_deltas,        // (N_ROIS, 4)
    float* __restrict__ out_wgts,          // (N_ROIS, 4)
    int n_rois)
{
    __shared__ float4 s_gt4[N_GT];     // 16 KB
    __shared__ float4 s_meta[N_GT];    // 16 KB

    const int tid = threadIdx.x;

    // ---- Stage GT boxes / labels / batch ids into LDS (async DMA path) ----
    for (int g = tid; g < N_GT; g += BLOCK) {
        const float* src = gt_boxes + g * 5;
        float* dstb = (float*)&s_gt4[g];
        async_g2l_b32(dstb + 0, src + 0);
        async_g2l_b32(dstb + 1, src + 1);
        async_g2l_b32(dstb + 2, src + 2);
        async_g2l_b32(dstb + 3, src + 3);
        float* dstm = (float*)&s_meta[g];
        async_g2l_b32(dstm + 1, src + 4);            // label
        async_g2l_b32(dstm + 2, gt_binds + g);       // batch id (int bits)
    }
    wait_async0();
    __syncthreads();

    // ---- Precompute GT areas (loop-invariant across all ROIs) ----
    for (int g = tid; g < N_GT; g += BLOCK) {
        float4 b = s_gt4[g];
        s_meta[g].x = (b.z - b.x + 1.0f) * (b.w - b.y + 1.0f);
    }
    __syncthreads();

    const int r = blockIdx.x * BLOCK + tid;
    if (r >= n_rois) return;

    const float rx1 = rois[r * 5 + 0];
    const float ry1 = rois[r * 5 + 1];
    const float rx2 = rois[r * 5 + 2];
    const float ry2 = rois[r * 5 + 3];
    const int   rb  = roi_binds[r];

    const float ew = rx2 - rx1 + 1.0f;
    const float eh = ry2 - ry1 + 1.0f;
    const float area_r = ew * eh;

    float best = -1.0f;   // matches reference's -1 fill for other-batch pairs
    int   bi   = 0;       // argmax -> first max index

    // All 32 lanes read the same LDS address each iteration: broadcast, no
    // bank conflicts. Unroll to overlap ds_load latency with VALU.
    #pragma unroll 4
    for (int g = 0; g < N_GT; ++g) {
        const float4 b = s_gt4[g];
        const float4 m = s_meta[g];
        float ix1 = fmaxf(rx1, b.x);
        float iy1 = fmaxf(ry1, b.y);
        float ix2 = fminf(rx2, b.z);
        float iy2 = fminf(ry2, b.w);
        float iw  = fmaxf(ix2 - ix1 + 1.0f, 0.0f);
        float ih  = fmaxf(iy2 - iy1 + 1.0f, 0.0f);
        float inter = iw * ih;
        float den   = area_r + m.x - inter;
        float iou   = inter * __builtin_amdgcn_rcpf(den);
        bool ok = (__float_as_int(m.z) == rb) && (iou > best);
        best = ok ? iou : best;
        bi   = ok ? g   : bi;
    }

    // ---- Emit labels / deltas / weights ----
    const bool fg = (best >= 0.5f);
    const float4 g4 = s_gt4[bi];
    const float  gl = s_meta[bi].y;

    float gw  = g4.z - g4.x + 1.0f;
    float gh  = g4.w - g4.y + 1.0f;
    float gcx = g4.x + 0.5f * gw;
    float gcy = g4.y + 0.5f * gh;
    float ecx = rx1 + 0.5f * ew;
    float ecy = ry1 + 0.5f * eh;

    float d0 = (gcx - ecx) * __builtin_amdgcn_rcpf(ew);
    float d1 = (gcy - ecy) * __builtin_amdgcn_rcpf(eh);
    float d2 = __logf(gw * __builtin_amdgcn_rcpf(ew));
    float d3 = __logf(gh * __builtin_amdgcn_rcpf(eh));

    out_labels[r] = fg ? gl : 0.0f;  // int32 labels, exactly representable

    const float w = fg ? 1.0f : 0.0f;
    float4 dd;
    dd.x = fg ? d0 : 0.0f;
    dd.y = fg ? d1 : 0.0f;
    dd.z = fg ? d2 : 0.0f;
    dd.w = fg ? d3 : 0.0f;
    *(float4*)(out_deltas + 4 * r) = dd;                       // global_store_b128
    *(float4*)(out_wgts   + 4 * r) = make_float4(w, w, w, w);  // global_store_b128
}

extern "C" void kernel_launch(void* const* d_in, const int* in_sizes, int n_in,
                              void* d_out, int out_size, void* d_ws, size_t ws_size,
                              hipStream_t stream) {
    const float* rois      = (const float*)d_in[0];
    const int*   roi_binds = (const int*)  d_in[1];
    const float* gt_boxes  = (const float*)d_in[2];
    const int*   gt_binds  = (const int*)  d_in[3];

    const int n_rois = in_sizes[1];          // roi_batch_inds element count

    float* out       = (float*)d_out;        // [labels | deltas | bbwgts]
    float* out_lab   = out;
    float* out_delta = out + n_rois;
    float* out_wgt   = out + n_rois + 4 * n_rois;

    const int blocks = (n_rois + BLOCK - 1) / BLOCK;
    RoITarget_kernel<<<blocks, BLOCK, 0, stream>>>(
        rois, roi_binds, gt_boxes, gt_binds,
        out_lab, out_delta, out_wgt, n_rois);
}